// SaliencyAttnPool_32942399160396
// MI455X (gfx1250) — compile-verified
//
#include <hip/hip_runtime.h>
#include <math.h>

// Problem constants from the reference: B=64, T=1024, C=768.
#define B_DIM 64
#define T_DIM 1024
#define C_DIM 768

typedef __attribute__((ext_vector_type(2))) float v2f;
typedef __attribute__((ext_vector_type(8))) float v8f;

// ---------------------------------------------------------------------------
// Kernel 1: q_sal[b,c] = cnt>0 ? sum_t m[b,t]*tok[b,t,c]/cnt : mean_t tok[b,t,c]
// Grid: (B, C/256), block 256. Mask staged through LDS; token reads coalesced
// (consecutive threads -> consecutive channels).
// ---------------------------------------------------------------------------
__global__ void k_qsal(const float* __restrict__ tokens,
                       const int* __restrict__ mask,
                       float* __restrict__ q_sal) {
    const int b = blockIdx.x;
    const int c = blockIdx.y * 256 + threadIdx.x;
    const int tid = threadIdx.x;

    __shared__ float sm[T_DIM];
    __shared__ float red[256];

    float cnt_part = 0.0f;
    for (int t = tid; t < T_DIM; t += 256) {
        float mv = (mask[b * T_DIM + t] != 0) ? 1.0f : 0.0f;
        sm[t] = mv;
        cnt_part += mv;
    }
    red[tid] = cnt_part;
    __syncthreads();
    for (int s = 128; s > 0; s >>= 1) {
        if (tid < s) red[tid] += red[tid + s];
        __syncthreads();
    }
    const float cnt = red[0];
    __syncthreads();

    const float* tp = tokens + (size_t)b * T_DIM * C_DIM + c;
    float s_masked = 0.0f, s_all = 0.0f;
    for (int t = 0; t < T_DIM; ++t) {
        float v = tp[(size_t)t * C_DIM];
        s_all += v;
        s_masked = fmaf(v, sm[t], s_masked);
    }
    float q = (cnt > 0.0f) ? (s_masked / fmaxf(cnt, 1.0f))
                           : (s_all * (1.0f / (float)T_DIM));
    q_sal[b * C_DIM + c] = q;
}

// ---------------------------------------------------------------------------
// Kernel 2: fp32 WMMA GEMM  Cout[M=64, N=768] = A[64,768] * Bmat[768,768]
//   TRANS_W==false : Bmat[k][n] = W[k*N + n]      (kq = Q @ Wk)
//   TRANS_W==true  : Bmat[k][n] = W[n*K + k]      (X @ W^T for Wq / Wv)
// TRANS_W is a template parameter so the inner K-loop is branch-free
// straight-line code: loads -> v_wmma_f32_16x16x4_f32, no scalar cbranch
// ladder between them (round-1 asm showed per-iteration branches when this
// was a runtime flag).
// One wave (32 threads) per 16x16 tile, stepping K by 4 using
// V_WMMA_F32_16X16X4_F32 (exact fp32 -> matches reference math).
// A-frag (16x4): lane L -> M=L%16, VGPR v -> K = v + 2*(L>=16).
// B-frag (4x16): lane L -> N=L%16, VGPR v -> K = v + 2*(L>=16).
// C/D (16x16):   lane L -> N=L%16, VGPR v -> M = v + 8*(L>=16).
// ---------------------------------------------------------------------------
template <bool TRANS_W>
__global__ void gemm_wmma_f32(const float* __restrict__ A,
                              const float* __restrict__ W,
                              float* __restrict__ Cout) {
    const int N = C_DIM, K = C_DIM;
    const int n0 = blockIdx.x * 16;
    const int m0 = blockIdx.y * 16;
    const int lane = threadIdx.x;           // 0..31, full wave, EXEC all 1s
    const int row  = lane & 15;
    const int kh   = (lane >> 4) * 2;       // 0 or 2

    v8f acc = {};
    const float* arow = A + (size_t)(m0 + row) * K;
    const float* wrow = TRANS_W ? (W + (size_t)(n0 + row) * K) : W;

    for (int k0 = 0; k0 < K; k0 += 4) {
        v2f af, bf;
        af.x = arow[k0 + kh];
        af.y = arow[k0 + kh + 1];
        if (TRANS_W) {
            bf.x = wrow[k0 + kh];
            bf.y = wrow[k0 + kh + 1];
        } else {
            bf.x = wrow[(size_t)(k0 + kh) * N + n0 + row];
            bf.y = wrow[(size_t)(k0 + kh + 1) * N + n0 + row];
        }
        acc = __builtin_amdgcn_wmma_f32_16x16x4_f32(
            /*neg_a=*/false, af, /*neg_b=*/false, bf,
            /*c_mod=*/(short)0, acc, /*reuse_a=*/false, /*reuse_b=*/false);
    }

    const int mo = (lane >= 16) ? 8 : 0;
#pragma unroll
    for (int v = 0; v < 8; ++v) {
        Cout[(size_t)(m0 + v + mo) * N + n0 + row] = acc[v];
    }
}

// ---------------------------------------------------------------------------
// Kernel 3: per-batch fused scores -> softmax -> attn output -> pooled.
// scores[t] = (tokens[b,t,:] . kq[b,:]) / sqrt(C)   (wave-per-token, lane=channel)
// attn = softmax(scores); pooled[c] = sum_t attn[t]*tokens[b,t,c]
// Grid: B blocks of 256 threads (8 waves).
// ---------------------------------------------------------------------------
__global__ void k_attn(const float* __restrict__ tokens,
                       const float* __restrict__ kq,
                       float* __restrict__ attn_out,
                       float* __restrict__ pooled) {
    const int b = blockIdx.x;
    const int tid = threadIdx.x;
    const int wave = tid >> 5;
    const int lane = tid & 31;

    __shared__ float skq[C_DIM];
    __shared__ float ss[T_DIM];
    __shared__ float red[256];

    for (int c = tid; c < C_DIM; c += 256) skq[c] = kq[b * C_DIM + c];
    __syncthreads();

    const float scale = 1.0f / sqrtf((float)C_DIM);
    const float* tbase = tokens + (size_t)b * T_DIM * C_DIM;

    // scores: 8 waves, each handles T/8 tokens; lanes stride channels (coalesced)
    for (int t = wave; t < T_DIM; t += 8) {
        const float* tp = tbase + (size_t)t * C_DIM;
        float acc = 0.0f;
        for (int c = lane; c < C_DIM; c += 32) acc = fmaf(tp[c], skq[c], acc);
        for (int off = 16; off > 0; off >>= 1) acc += __shfl_xor(acc, off, 32);
        if (lane == 0) ss[t] = acc * scale;
    }
    __syncthreads();

    // block max
    float m = -3.402823466e38f;
    for (int i = tid; i < T_DIM; i += 256) m = fmaxf(m, ss[i]);
    red[tid] = m;
    __syncthreads();
    for (int s = 128; s > 0; s >>= 1) {
        if (tid < s) red[tid] = fmaxf(red[tid], red[tid + s]);
        __syncthreads();
    }
    const float smax = red[0];
    __syncthreads();

    // exp + sum
    float sum = 0.0f;
    for (int i = tid; i < T_DIM; i += 256) {
        float e = expf(ss[i] - smax);
        ss[i] = e;
        sum += e;
    }
    red[tid] = sum;
    __syncthreads();
    for (int s = 128; s > 0; s >>= 1) {
        if (tid < s) red[tid] += red[tid + s];
        __syncthreads();
    }
    const float inv = 1.0f / red[0];
    __syncthreads();

    for (int i = tid; i < T_DIM; i += 256) {
        float a = ss[i] * inv;
        ss[i] = a;
        attn_out[b * T_DIM + i] = a;
    }
    __syncthreads();

    // pooled[c] = sum_t attn[t] * tokens[b,t,c]  (threads stride channels)
    for (int c = tid; c < C_DIM; c += 256) {
        const float* tp = tbase + c;
        float acc = 0.0f;
        for (int t = 0; t < T_DIM; ++t) acc = fmaf(ss[t], tp[(size_t)t * C_DIM], acc);
        pooled[b * C_DIM + c] = acc;
    }
}

// ---------------------------------------------------------------------------
// Host launcher
// ---------------------------------------------------------------------------
extern "C" void kernel_launch(void* const* d_in, const int* in_sizes, int n_in,
                              void* d_out, int out_size, void* d_ws, size_t ws_size,
                              hipStream_t stream) {
    (void)in_sizes; (void)n_in; (void)out_size; (void)ws_size;

    const float* tokens = (const float*)d_in[0];
    const int*   mask   = (const int*)d_in[1];
    const float* Wq     = (const float*)d_in[2];
    const float* Wk     = (const float*)d_in[3];
    const float* Wv     = (const float*)d_in[4];

    float* out  = (float*)d_out;
    float* v_fg = out;                               // 64*768
    float* attn = out + B_DIM * C_DIM;               // 64*1024

    float* ws     = (float*)d_ws;
    float* q_sal  = ws;                              // 64*768
    float* Qm     = ws + 1 * B_DIM * C_DIM;          // 64*768
    float* kq     = ws + 2 * B_DIM * C_DIM;          // 64*768
    float* pooled = ws + 3 * B_DIM * C_DIM;          // 64*768

    const dim3 gemm_grid(C_DIM / 16, B_DIM / 16);

    // 1) masked-mean query
    k_qsal<<<dim3(B_DIM, C_DIM / 256), 256, 0, stream>>>(tokens, mask, q_sal);

    // 2) Q = q_sal @ Wq^T   (WMMA fp32, trans)
    gemm_wmma_f32<true><<<gemm_grid, 32, 0, stream>>>(q_sal, Wq, Qm);

    // 3) kq = Q @ Wk        (WMMA fp32, normal)
    gemm_wmma_f32<false><<<gemm_grid, 32, 0, stream>>>(Qm, Wk, kq);

    // 4) scores/softmax/attn/pooled fused, one block per batch
    k_attn<<<B_DIM, 256, 0, stream>>>(tokens, kq, attn, pooled);

    // 5) v_fg = pooled @ Wv^T (WMMA fp32, trans)
    gemm_wmma_f32<true><<<gemm_grid, 32, 0, stream>>>(pooled, Wv, v_fg);
}